// SetenceDecoder_77695958385129
// MI455X (gfx1250) — compile-verified
//
#include <hip/hip_runtime.h>
#include <hip/hip_bf16.h>
#include <math.h>

// Problem dims (fixed by the reference)
#define VOCAB_N 10000
#define IMG_E   1024
#define EMB_N   512
#define HID_N   512
#define P_N     49
#define B_N     64
#define T_N     50

typedef __attribute__((ext_vector_type(2))) float v2f;
typedef __attribute__((ext_vector_type(8))) float v8f;

// ---------------------------------------------------------------------------
// fp32 WMMA primitive: D(16x16) += A(16x4) * B(4x16), full fp32 on the matrix
// pipe (V_WMMA_F32_16X16X4_F32). 8-arg form:
// (neg_a, A, neg_b, B, c_mod, C, reuse_a, reuse_b)
// ---------------------------------------------------------------------------
__device__ __forceinline__ v8f wmma4(v2f a, v2f b, v8f c) {
  return __builtin_amdgcn_wmma_f32_16x16x4_f32(
      false, a, false, b, (short)0, c, false, false);
}

// ---------------------------------------------------------------------------
// Multi-tile K-segment: one wave computes NT adjacent 16x16 output tiles
// (16 x 16*NT). A is loaded once per K-step and reused across all NT tiles;
// 2*NT independent accumulator chains keep the WMMA pipe busy.
// arow: per-lane A pointer already offset by +2*half (lane-half gives K=2,3).
// bbase: B pointer already offset by (2*half)*ldb + n0 + l16.
// ---------------------------------------------------------------------------
template <int NT>
__device__ __forceinline__ void gemm_seg_nt(const float* __restrict__ arow,
                                            const float* __restrict__ bbase,
                                            int kbase, int klen, int ldb,
                                            v8f (&acc0)[NT], v8f (&acc1)[NT]) {
  for (int kk = 0; kk < klen; kk += 8) {
    const int k = kbase + kk;
    v2f a0, a1;
    a0.x = arow[kk];     a0.y = arow[kk + 1];
    a1.x = arow[kk + 4]; a1.y = arow[kk + 5];
    // prefetch next K-iteration of the streaming B operand (tile 0 row)
    if (kk + 8 < klen)
      __builtin_prefetch(bbase + (size_t)(k + 8) * ldb, 0, 1);
#pragma unroll
    for (int tn = 0; tn < NT; ++tn) {
      const float* bc = bbase + 16 * tn + (size_t)k * ldb;
      v2f b0, b1;
      b0.x = bc[0];
      b0.y = bc[(size_t)ldb];
      b1.x = bc[(size_t)4 * ldb];
      b1.y = bc[(size_t)5 * ldb];
      acc0[tn] = wmma4(a0, b0, acc0[tn]);
      acc1[tn] = wmma4(a1, b1, acc1[tn]);
    }
  }
}

// ---------------------------------------------------------------------------
// Generic row-major GEMM  C[M,N] = A[M,K] @ B[K,N] + bias[N]
// One wave per 16 x (16*NT) tile strip. grid = (N/(16*NT), M/16), block = 32.
// N must be a multiple of 16*NT (true for all launches below).
// ---------------------------------------------------------------------------
template <int NT>
__global__ void gemm_wmma_f32(const float* __restrict__ A,
                              const float* __restrict__ B,
                              const float* __restrict__ bias,
                              float* __restrict__ C,
                              int M, int N, int K, int lda, int ldb, int ldc) {
  const int lane = threadIdx.x & 31;
  const int half = lane >> 4;
  const int l16  = lane & 15;
  const int m0 = blockIdx.y * 16;
  const int n0 = blockIdx.x * (16 * NT);

  const float* arow  = A + (size_t)(m0 + l16) * lda + 2 * half;
  const float* bbase = B + (size_t)(2 * half) * ldb + n0 + l16;

  v8f acc0[NT], acc1[NT];
  const v8f vzero = {};
#pragma unroll
  for (int tn = 0; tn < NT; ++tn) { acc0[tn] = vzero; acc1[tn] = vzero; }

  gemm_seg_nt<NT>(arow, bbase, 0, K, ldb, acc0, acc1);

#pragma unroll
  for (int tn = 0; tn < NT; ++tn) {
    const int col = n0 + 16 * tn + l16;
    const float bb = bias ? bias[col] : 0.0f;
#pragma unroll
    for (int r = 0; r < 8; ++r) {
      const int row = m0 + r + 8 * half;
      if (row < M)
        C[(size_t)row * ldc + col] = acc0[tn][r] + acc1[tn][r] + bb;
    }
  }
}

// ---------------------------------------------------------------------------
// Gates GEMM with gathered A:  x = [emb[w[b,t]], vg[b], h[b]]  (K = 1536)
// gates[64,2560] = x @ Wl + bl.  NT=5 -> grid = (32, 4), block = 32.
// ---------------------------------------------------------------------------
__global__ void gates_gemm(const int* __restrict__ w,
                           const float* __restrict__ emb,
                           const float* __restrict__ vg,
                           const float* __restrict__ h,
                           const float* __restrict__ Wl,
                           const float* __restrict__ bl,
                           float* __restrict__ gates, int t) {
  constexpr int NT = 5;
  const int lane = threadIdx.x & 31;
  const int half = lane >> 4;
  const int l16  = lane & 15;
  const int m0 = blockIdx.y * 16;
  const int n0 = blockIdx.x * (16 * NT);
  const int b  = m0 + l16;                 // batch row handled by this lane

  const float* embrow = emb + (size_t)w[b * T_N + t] * EMB_N + 2 * half;
  const float* vgrow  = vg + (size_t)b * EMB_N + 2 * half;
  const float* hrow   = h  + (size_t)b * HID_N + 2 * half;
  const float* bbase  = Wl + (size_t)(2 * half) * 2560 + n0 + l16;

  v8f acc0[NT], acc1[NT];
  const v8f vzero = {};
#pragma unroll
  for (int tn = 0; tn < NT; ++tn) { acc0[tn] = vzero; acc1[tn] = vzero; }

  gemm_seg_nt<NT>(embrow, bbase,    0, EMB_N, 2560, acc0, acc1);
  gemm_seg_nt<NT>(vgrow,  bbase,  512, EMB_N, 2560, acc0, acc1);
  gemm_seg_nt<NT>(hrow,   bbase, 1024, HID_N, 2560, acc0, acc1);

#pragma unroll
  for (int tn = 0; tn < NT; ++tn) {
    const int col = n0 + 16 * tn + l16;
    const float bb = bl[col];
#pragma unroll
    for (int r = 0; r < 8; ++r) {
      const int row = m0 + r + 8 * half;
      gates[(size_t)row * 2560 + col] = acc0[tn][r] + acc1[tn][r] + bb;
    }
  }
}

// ---------------------------------------------------------------------------
// LSTM elementwise: split gates -> i,f,o,g,m5; update m, produce h, s.
// 64*512 threads total.
// ---------------------------------------------------------------------------
__global__ void lstm_elem(const float* __restrict__ gates,
                          float* __restrict__ h, float* __restrict__ m,
                          float* __restrict__ s) {
  const int idx = blockIdx.x * blockDim.x + threadIdx.x;  // 0 .. 64*512-1
  const int b = idx >> 9;
  const int j = idx & 511;
  const float* g = gates + (size_t)b * 2560;
  const float iv = 1.0f / (1.0f + __expf(-g[j]));
  const float fv = 1.0f / (1.0f + __expf(-g[512 + j]));
  const float ov = 1.0f / (1.0f + __expf(-g[1024 + j]));
  const float gv = g[1536 + j];                 // sentinel gate: raw (as in ref)
  const float m5 = g[2048 + j];
  const float mn = iv * tanhf(m5) + fv * m[idx];
  const float tm = tanhf(mn);
  m[idx] = mn;
  h[idx] = ov * tm;
  s[idx] = gv * tm;
}

// ---------------------------------------------------------------------------
// Adaptive attention + context.  One block per batch row, 256 threads.
// scores p<49: Waz . tanh(pv[b,p]+ps[b]) ; p==49 (sentinel): tanh(ph+ps)
// softmax over 50, c = sum a_p * vproj[b,p] + a_49 * h ; ch = c + h.
// ---------------------------------------------------------------------------
__global__ void attn_ctx(const float* __restrict__ pv,
                         const float* __restrict__ vproj,
                         const float* __restrict__ h,
                         const float* __restrict__ ph,
                         const float* __restrict__ ps,
                         const float* __restrict__ Waz,
                         const float* __restrict__ baz,
                         float* __restrict__ ch) {
  __shared__ float s_ps[HID_N];
  __shared__ float s_ph[HID_N];
  __shared__ float s_sc[64];
  const int b   = blockIdx.x;
  const int tid = threadIdx.x;                  // 256 threads = 8 waves
  for (int j = tid; j < HID_N; j += 256) {
    s_ps[j] = ps[(size_t)b * HID_N + j];
    s_ph[j] = ph[(size_t)b * HID_N + j];
  }
  __syncthreads();

  const int wid = tid >> 5, lane = tid & 31;
  for (int p = wid; p < P_N + 1; p += 8) {
    float part = 0.0f;
    if (p < P_N) {
      const float* src = pv + ((size_t)b * P_N + p) * HID_N;
      for (int j = lane; j < HID_N; j += 32)
        part += Waz[j] * tanhf(src[j] + s_ps[j]);
    } else {
      for (int j = lane; j < HID_N; j += 32)
        part += Waz[j] * tanhf(s_ph[j] + s_ps[j]);
    }
    for (int off = 16; off; off >>= 1) part += __shfl_xor(part, off, 32);
    if (lane == 0) s_sc[p] = part + baz[0];
  }
  __syncthreads();

  if (tid == 0) {                               // 50-way softmax: trivial
    float mx = s_sc[0];
    for (int p = 1; p < P_N + 1; ++p) mx = fmaxf(mx, s_sc[p]);
    float sum = 0.0f;
    for (int p = 0; p < P_N + 1; ++p) {
      const float e = __expf(s_sc[p] - mx);
      s_sc[p] = e; sum += e;
    }
    const float inv = 1.0f / sum;
    for (int p = 0; p < P_N + 1; ++p) s_sc[p] *= inv;
  }
  __syncthreads();

  for (int j = tid; j < HID_N; j += 256) {
    const float hv = h[(size_t)b * HID_N + j];
    float c = s_sc[P_N] * hv;
    for (int p = 0; p < P_N; ++p)
      c += s_sc[p] * vproj[((size_t)b * P_N + p) * HID_N + j];
    ch[(size_t)b * HID_N + j] = c + hv;
  }
}

// ---------------------------------------------------------------------------
// Vocab softmax + length mask -> out[b, t, :].  One block per batch row.
// ---------------------------------------------------------------------------
__global__ void vocab_softmax(const float* __restrict__ logits,
                              const int* __restrict__ lengths,
                              float* __restrict__ out, int t) {
  __shared__ float red[8];
  const int b = blockIdx.x;
  const int tid = threadIdx.x;                  // 256
  const float* row = logits + (size_t)b * VOCAB_N;

  float mx = -3.4e38f;
  for (int j = tid; j < VOCAB_N; j += 256) mx = fmaxf(mx, row[j]);
  for (int off = 16; off; off >>= 1) mx = fmaxf(mx, __shfl_xor(mx, off, 32));
  if ((tid & 31) == 0) red[tid >> 5] = mx;
  __syncthreads();
  if (tid == 0) {
    float v = red[0];
    for (int i = 1; i < 8; ++i) v = fmaxf(v, red[i]);
    red[0] = v;
  }
  __syncthreads();
  mx = red[0];
  __syncthreads();

  float sum = 0.0f;
  for (int j = tid; j < VOCAB_N; j += 256) sum += __expf(row[j] - mx);
  for (int off = 16; off; off >>= 1) sum += __shfl_xor(sum, off, 32);
  if ((tid & 31) == 0) red[tid >> 5] = sum;
  __syncthreads();
  if (tid == 0) {
    float v = 0.0f;
    for (int i = 0; i < 8; ++i) v += red[i];
    red[0] = v;
  }
  __syncthreads();
  sum = red[0];

  const float inv = (t < lengths[b]) ? (1.0f / sum) : 0.0f;
  float* o = out + ((size_t)b * T_N + t) * VOCAB_N;
  for (int j = tid; j < VOCAB_N; j += 256) o[j] = __expf(row[j] - mx) * inv;
}

// ---------------------------------------------------------------------------
// vg = mean over P of vproj
// ---------------------------------------------------------------------------
__global__ void vg_mean(const float* __restrict__ vproj, float* __restrict__ vg) {
  const int b = blockIdx.x;
  const int tid = threadIdx.x;
  for (int j = tid; j < EMB_N; j += 256) {
    float a = 0.0f;
    for (int p = 0; p < P_N; ++p)
      a += vproj[((size_t)b * P_N + p) * EMB_N + j];
    vg[(size_t)b * EMB_N + j] = a * (1.0f / (float)P_N);
  }
}

// ---------------------------------------------------------------------------
extern "C" void kernel_launch(void* const* d_in, const int* in_sizes, int n_in,
                              void* d_out, int out_size, void* d_ws, size_t ws_size,
                              hipStream_t stream) {
  const float* v      = (const float*)d_in[0];   // [64,49,1024]
  const int*   w      = (const int*)  d_in[1];   // [64,50]
  const int*   lens   = (const int*)  d_in[2];   // [64]
  const float* emb    = (const float*)d_in[3];   // [10000,512]
  const float* Wv     = (const float*)d_in[4];
  const float* bv     = (const float*)d_in[5];
  const float* Wh     = (const float*)d_in[6];
  const float* bh     = (const float*)d_in[7];
  const float* Wm     = (const float*)d_in[8];
  const float* bm     = (const float*)d_in[9];
  const float* Wl     = (const float*)d_in[10];  // [1536,2560]
  const float* bl     = (const float*)d_in[11];
  const float* Wav    = (const float*)d_in[12];
  const float* bav    = (const float*)d_in[13];
  const float* Wah    = (const float*)d_in[14];
  const float* bah    = (const float*)d_in[15];
  const float* Was    = (const float*)d_in[16];
  const float* bas    = (const float*)d_in[17];
  const float* Waz    = (const float*)d_in[18];  // [512,1]
  const float* baz    = (const float*)d_in[19];  // [1]
  const float* Wp     = (const float*)d_in[20];  // [512,10000]
  const float* bp     = (const float*)d_in[21];
  float* out = (float*)d_out;                    // [64,50,10000] fp32

  // Workspace layout (floats), ~17 MB total
  float* ws = (float*)d_ws;
  float* vproj  = ws;                      // 3136*512
  float* pvb    = vproj  + 3136 * 512;     // 3136*512
  float* vg     = pvb    + 3136 * 512;     // 64*512
  float* hbuf   = vg     + 64 * 512;       // 64*512
  float* mbuf   = hbuf   + 64 * 512;       // 64*512
  float* sbuf   = mbuf   + 64 * 512;       // 64*512
  float* phbuf  = sbuf   + 64 * 512;       // 64*512
  float* psbuf  = phbuf  + 64 * 512;       // 64*512
  float* chbuf  = psbuf  + 64 * 512;       // 64*512
  float* gates  = chbuf  + 64 * 512;       // 64*2560
  float* logits = gates  + 64 * 2560;      // 64*10000

  const dim3 wave(32);

  // ---- Precompute (once per call) ----
  // vproj = v @ Wv + bv : [3136,1024]x[1024,512]; 16x64 per wave
  gemm_wmma_f32<4><<<dim3(EMB_N / 64, (B_N * P_N) / 16), wave, 0, stream>>>(
      v, Wv, bv, vproj, B_N * P_N, EMB_N, IMG_E, IMG_E, EMB_N, EMB_N);
  // vg = mean_P(vproj)
  vg_mean<<<B_N, 256, 0, stream>>>(vproj, vg);
  // h0 = vg @ Wh + bh ; m0 = vg @ Wm + bm
  gemm_wmma_f32<4><<<dim3(HID_N / 64, B_N / 16), wave, 0, stream>>>(
      vg, Wh, bh, hbuf, B_N, HID_N, EMB_N, EMB_N, HID_N, HID_N);
  gemm_wmma_f32<4><<<dim3(HID_N / 64, B_N / 16), wave, 0, stream>>>(
      vg, Wm, bm, mbuf, B_N, HID_N, EMB_N, EMB_N, HID_N, HID_N);
  // pv = vproj @ Wav + bav : [3136,512]x[512,512]
  gemm_wmma_f32<4><<<dim3(HID_N / 64, (B_N * P_N) / 16), wave, 0, stream>>>(
      vproj, Wav, bav, pvb, B_N * P_N, HID_N, EMB_N, EMB_N, HID_N, HID_N);

  // ---- Sequential scan over T timesteps (stream-ordered) ----
  for (int t = 0; t < T_N; ++t) {
    // gates = [emb[w[:,t]], vg, h] @ Wl + bl : [64,1536]x[1536,2560]; 16x80/wave
    gates_gemm<<<dim3(2560 / 80, B_N / 16), wave, 0, stream>>>(
        w, emb, vg, hbuf, Wl, bl, gates, t);
    // LSTM elementwise -> h, m, s
    lstm_elem<<<(B_N * HID_N) / 256, 256, 0, stream>>>(gates, hbuf, mbuf, sbuf);
    // ph = h @ Wah + bah ; ps = s @ Was + bas
    gemm_wmma_f32<4><<<dim3(HID_N / 64, B_N / 16), wave, 0, stream>>>(
        hbuf, Wah, bah, phbuf, B_N, HID_N, HID_N, HID_N, HID_N, HID_N);
    gemm_wmma_f32<4><<<dim3(HID_N / 64, B_N / 16), wave, 0, stream>>>(
        sbuf, Was, bas, psbuf, B_N, HID_N, HID_N, HID_N, HID_N, HID_N);
    // adaptive attention + context; ch = c + h
    attn_ctx<<<B_N, 256, 0, stream>>>(pvb, vproj, hbuf, phbuf, psbuf, Waz, baz,
                                      chbuf);
    // logits = ch @ Wp + bp : [64,512]x[512,10000]; 16x80 per wave
    gemm_wmma_f32<5><<<dim3(VOCAB_N / 80, B_N / 16), wave, 0, stream>>>(
        chbuf, Wp, bp, logits, B_N, VOCAB_N, HID_N, HID_N, VOCAB_N, VOCAB_N);
    // softmax + length mask -> out[:, t, :]
    vocab_softmax<<<B_N, 256, 0, stream>>>(logits, lens, out, t);
  }
}